// AggregationRebuild_HN_20882130993692
// MI455X (gfx1250) — compile-verified
//
#include <hip/hip_runtime.h>
#include <hip/hip_bf16.h>

typedef __attribute__((ext_vector_type(2))) float v2f;
typedef __attribute__((ext_vector_type(8))) float v8f;

#define TEMP_INV 50.0f     // 1 / 0.02
#define B_ROWS   512       // bs * n_vars
#define KP       3
#define KN       16
#define KT       19        // KP + KN
#define KPAD     20        // padded to multiple of 4 for K=4 WMMA chunks
#define PD       8192      // P * D
#define NSEG     8
#define NWAVE    4
#define NITER    ((PD / NSEG) / (NWAVE * 32))   // 8 iterations of 32 cols/wave

// ---------------------------------------------------------------------------
// Kernel 1: temperature-scaled softmax over the 19 similarity scores per row.
// Writes padded weight rows w[b][0..19] (w[19] = 0) into workspace.
// ---------------------------------------------------------------------------
__global__ void softmax_weights(const float* __restrict__ sim,
                                float* __restrict__ w) {
    int b = blockIdx.x * blockDim.x + threadIdx.x;
    if (b >= B_ROWS) return;
    float v[KT];
    float m = -3.4e38f;
#pragma unroll
    for (int k = 0; k < KT; ++k) {
        v[k] = sim[b * KT + k] * TEMP_INV;
        m = fmaxf(m, v[k]);
    }
    float s = 0.f;
#pragma unroll
    for (int k = 0; k < KT; ++k) {
        v[k] = __expf(v[k] - m);
        s += v[k];
    }
    float inv = 1.0f / s;
#pragma unroll
    for (int k = 0; k < KT; ++k) w[b * KPAD + k] = v[k] * inv;
    w[b * KPAD + KT] = 0.f;   // padding weight for K-chunk 4 (k=19)
}

// ---------------------------------------------------------------------------
// Kernel 2: weighted gather-aggregation via V_WMMA_F32_16X16X4_F32.
// Per sample b: out[b, :] = sum_{k<19} w[b,k] * src_k[:], 5 chained K=4 WMMAs.
// A(16x4): weights broadcast over M.  Layout per ISA (A 16x4, 32-bit):
//   VGPR0: lanes 0-15 K=0 | lanes 16-31 K=2 ; VGPR1: K=1 | K=3
// B(4x16): row K striped over 16 lanes (N), same K ordering as A.
// Two adjacent 16-column blocks per iteration -> full-wave coalesced store.
// Column loop fully unrolled: per-iteration offsets (it*512 B, max 3648 B)
// fold into load/store 24-bit immediates -> zero per-iteration address VALU.
// ---------------------------------------------------------------------------
__global__ __launch_bounds__(NWAVE * 32)
void agg_rebuild_wmma(const float* __restrict__ penc,
                      const int*   __restrict__ negidx,
                      const float* __restrict__ wts,
                      float*       __restrict__ out) {
    const int b    = blockIdx.x;
    const int lane = threadIdx.x & 31;
    const int wave = threadIdx.x >> 5;
    const int hi   = (lane >> 4) & 1;     // which half-wave (K+0/1 vs K+2/3)

    const float* wrow = wts + b * KPAD;

    // Per-lane source-row base pointers and A-operand registers (built once).
    const float* srcp[10];
    v2f areg[5];
#pragma unroll
    for (int c = 0; c < 5; ++c) {
#pragma unroll
        for (int e = 0; e < 2; ++e) {
            int   k = 4 * c + 2 * hi + e;   // k in 0..19
            int   row;
            float wk;
            if (k < KP)      { row = B_ROWS + b * KP + k;        wk = wrow[k]; }
            else if (k < KT) { row = negidx[b * KN + (k - KP)];  wk = wrow[k]; }
            else             { row = 0;                          wk = 0.f; }
            srcp[2 * c + e] = penc + (size_t)row * PD;
            if (e == 0) areg[c].x = wk; else areg[c].y = wk;
        }
    }

    float* orow = out + (size_t)b * PD;
    const int segbase = blockIdx.y * (PD / NSEG);
    const int col0    = segbase + wave * 32;       // iteration 0 column base
    const int n0      = col0 + (lane & 15);        // per-lane B-operand column

#pragma unroll
    for (int it = 0; it < NITER; ++it) {
        const int coff = it * (NWAVE * 32);        // compile-time constant

        v8f acc  = {0.f, 0.f, 0.f, 0.f, 0.f, 0.f, 0.f, 0.f};
        v8f acc2 = {0.f, 0.f, 0.f, 0.f, 0.f, 0.f, 0.f, 0.f};
#pragma unroll
        for (int c = 0; c < 5; ++c) {
            v2f b1, b2;
            b1.x = srcp[2 * c + 0][n0 + coff];
            b1.y = srcp[2 * c + 1][n0 + coff];
            b2.x = srcp[2 * c + 0][n0 + coff + 16];
            b2.y = srcp[2 * c + 1][n0 + coff + 16];
            acc  = __builtin_amdgcn_wmma_f32_16x16x4_f32(
                       false, areg[c], false, b1, (short)0, acc,  false, false);
            acc2 = __builtin_amdgcn_wmma_f32_16x16x4_f32(
                       false, areg[c], false, b2, (short)0, acc2, false, false);
        }
        // All C/D rows are identical (A rows identical); element 0 of each
        // half-wave holds the correct column value for its block.
        float r = (lane < 16) ? acc[0] : acc2[0];
        orow[col0 + coff + lane] = r;
    }
}

// ---------------------------------------------------------------------------
extern "C" void kernel_launch(void* const* d_in, const int* in_sizes, int n_in,
                              void* d_out, int out_size, void* d_ws, size_t ws_size,
                              hipStream_t stream) {
    const float* sim    = (const float*)d_in[0];   // [512, 19] f32
    const float* penc   = (const float*)d_in[1];   // [2048, 64, 128] f32
    const int*   negidx = (const int*)  d_in[2];   // [512, 16] i32
    float*       out    = (float*)d_out;           // [512, 64, 128] f32
    float*       w      = (float*)d_ws;            // [512, 20] f32 scratch

    hipLaunchKernelGGL(softmax_weights, dim3(2), dim3(256), 0, stream, sim, w);
    hipLaunchKernelGGL(agg_rebuild_wmma, dim3(B_ROWS, NSEG), dim3(NWAVE * 32),
                       0, stream, penc, negidx, w, out);
}